// NonMaxSuppression_25366076850407
// MI455X (gfx1250) — compile-verified
//
#include <hip/hip_runtime.h>
#include <hip/hip_bf16.h>
#include <stdint.h>

// ---------------------------------------------------------------------------
// NMS (3x3 max-pool, stride 1) on (8,1,2048,2048) fp32.
// Bandwidth-bound: 134 MB in + 134 MB out => ~11.5 us floor @ 23.3 TB/s.
// Strategy: async DMA global->LDS staging (CDNA5 GLOBAL_LOAD_ASYNC_TO_LDS_*,
// ASYNCcnt), then separable 3x3 max from LDS, 128-bit coalesced I/O.
// ---------------------------------------------------------------------------

#define NMS_W 2048
#define NMS_H 2048
#define NMS_BORDER 10
#define NMS_THR 0.6f

#define TILE_W 1024       // columns per block
#define TILE_H 8          // output rows per block
#define STAGE_ROWS (TILE_H + 2)
#define LDSW (TILE_W + 8) // main data at [4..1027]; halo L at 3, halo R at 1028

// Types matching the builtin signatures leaked by the round-1 diagnostics:
//   b128: (v4i as1*, v4i as3*, Ii, Ii)   b32: (int as1*, int as3*, Ii, Ii)
typedef int v4i_vs __attribute__((vector_size(4 * sizeof(int))));
typedef __attribute__((address_space(1))) v4i_vs* g_v4i_ptr;
typedef __attribute__((address_space(3))) v4i_vs* l_v4i_ptr;
typedef __attribute__((address_space(1))) int*    g_i32_ptr;
typedef __attribute__((address_space(3))) int*    l_i32_ptr;

// ---- CDNA5 async global->LDS copy helpers ---------------------------------

__device__ __forceinline__ void async_copy_b128(const float* gsrc, float* ldst) {
#if __has_builtin(__builtin_amdgcn_global_load_async_to_lds_b128)
  // LDS generic VA: low 32 bits are the LDS byte offset (ISA 10.2 aperture).
  __builtin_amdgcn_global_load_async_to_lds_b128(
      (g_v4i_ptr)(size_t)gsrc,
      (l_v4i_ptr)(uint32_t)(size_t)ldst,
      0, 0);
#else
  uint32_t lds_addr = (uint32_t)(size_t)ldst;
  asm volatile("global_load_async_to_lds_b128 %0, %1, off"
               :: "v"(lds_addr), "v"(gsrc) : "memory");
#endif
}

__device__ __forceinline__ void async_copy_b32(const float* gsrc, float* ldst) {
#if __has_builtin(__builtin_amdgcn_global_load_async_to_lds_b32)
  __builtin_amdgcn_global_load_async_to_lds_b32(
      (g_i32_ptr)(size_t)gsrc,
      (l_i32_ptr)(uint32_t)(size_t)ldst,
      0, 0);
#else
  uint32_t lds_addr = (uint32_t)(size_t)ldst;
  asm volatile("global_load_async_to_lds_b32 %0, %1, off"
               :: "v"(lds_addr), "v"(gsrc) : "memory");
#endif
}

__device__ __forceinline__ void wait_async_zero() {
#if __has_builtin(__builtin_amdgcn_s_wait_asynccnt)
  __builtin_amdgcn_s_wait_asynccnt(0);
#else
  asm volatile("s_wait_asynccnt 0" ::: "memory");
#endif
}

__device__ __forceinline__ int clampi(int v, int lo, int hi) {
  return v < lo ? lo : (v > hi ? hi : v);
}

// ---------------------------------------------------------------------------

__global__ void __launch_bounds__(256)
nms_3x3_kernel(const float* __restrict__ in, float* __restrict__ out) {
  __shared__ float s[STAGE_ROWS * LDSW];

  const int t    = threadIdx.x;              // 0..255
  const int base = blockIdx.x * TILE_W;      // column base of tile
  const int y0   = blockIdx.y * TILE_H;      // first output row of tile
  const int b    = blockIdx.z;               // batch index

  const float* img = in + (size_t)b * NMS_H * NMS_W;

  // ---- Stage rows y0-1 .. y0+TILE_H into LDS via async DMA (b128/lane) ----
#pragma unroll
  for (int r = 0; r < STAGE_ROWS; ++r) {
    const int gy = clampi(y0 - 1 + r, 0, NMS_H - 1);   // clamped rows only
    const float* rowp = img + (size_t)gy * NMS_W;      // feed border pixels,
    async_copy_b128(rowp + base + 4 * t,               // which are masked out
                    &s[r * LDSW + 4 + 4 * t]);
  }
  // Column halos (1 float per staged row on each side), clamped at edges.
  if (t < STAGE_ROWS) {
    const int gy  = clampi(y0 - 1 + t, 0, NMS_H - 1);
    const int gxl = (base - 1 < 0) ? 0 : base - 1;
    async_copy_b32(img + (size_t)gy * NMS_W + gxl, &s[t * LDSW + 3]);
  } else if (t >= 32 && t < 32 + STAGE_ROWS) {
    const int r   = t - 32;
    const int gy  = clampi(y0 - 1 + r, 0, NMS_H - 1);
    int gxr = base + TILE_W;
    if (gxr > NMS_W - 1) gxr = NMS_W - 1;
    async_copy_b32(img + (size_t)gy * NMS_W + gxr, &s[r * LDSW + 4 + TILE_W]);
  }

  wait_async_zero();   // own async DMA complete
  __syncthreads();     // all lanes' DMA visible in LDS

  // ---- Compute: separable 3x3 max from LDS, 4 cols x TILE_H rows/thread ---
  const int c0  = 4 + 4 * t;     // LDS column of first output
  const int gx0 = base + 4 * t;  // global column of first output

#pragma unroll
  for (int ry = 0; ry < TILE_H; ++ry) {
    const int gy = y0 + ry;
    const float* r0 = &s[(ry + 0) * LDSW];  // row gy-1
    const float* r1 = &s[(ry + 1) * LDSW];  // row gy   (center)
    const float* r2 = &s[(ry + 2) * LDSW];  // row gy+1

    float v[6];
#pragma unroll
    for (int k = 0; k < 6; ++k) {
      const int c = c0 - 1 + k;
      v[k] = fmaxf(r0[c], fmaxf(r1[c], r2[c]));
    }

    const bool yin = (gy >= NMS_BORDER) && (gy < NMS_H - NMS_BORDER);
    float4 o;
    float* op = &o.x;
#pragma unroll
    for (int j = 0; j < 4; ++j) {
      const float center = r1[c0 + j];
      const float m = fmaxf(v[j], fmaxf(v[j + 1], v[j + 2]));
      const int gx = gx0 + j;
      const bool xin = (gx >= NMS_BORDER) && (gx < NMS_W - NMS_BORDER);
      op[j] = (yin && xin && (center >= NMS_THR) && (center == m)) ? 1.0f : 0.0f;
    }
    *(float4*)(out + ((size_t)b * NMS_H + gy) * NMS_W + gx0) = o;
  }
}

// ---------------------------------------------------------------------------

extern "C" void kernel_launch(void* const* d_in, const int* in_sizes, int n_in,
                              void* d_out, int out_size, void* d_ws, size_t ws_size,
                              hipStream_t stream) {
  (void)n_in; (void)out_size; (void)d_ws; (void)ws_size;
  const float* in = (const float*)d_in[0];
  float* out = (float*)d_out;

  const int total = in_sizes[0];
  const int batch = total / (NMS_H * NMS_W);   // 8 for the reference shapes

  dim3 grid(NMS_W / TILE_W, NMS_H / TILE_H, batch);
  dim3 block(256, 1, 1);
  nms_3x3_kernel<<<grid, block, 0, stream>>>(in, out);
}